// MolDis_37331855737526
// MI455X (gfx1250) — compile-verified
//
#include <hip/hip_runtime.h>
#include <hip/hip_bf16.h>

#define N_NODES 524288
#define FEAT 32
#define N_REL 4

typedef __attribute__((ext_vector_type(2))) float v2f;
typedef __attribute__((ext_vector_type(8))) float v8f;

// ---------------- zero fill (ws is poisoned by harness) ----------------
__global__ __launch_bounds__(256) void zero_f4(float4* __restrict__ p, size_t n4) {
    size_t i = (size_t)blockIdx.x * blockDim.x + threadIdx.x;
    if (i < n4) p[i] = make_float4(0.f, 0.f, 0.f, 0.f);
}

// ---------------- per-relation scatter-add of raw features ----------------
// S[r][dst][f] += feat[src][f]   (one wave == one edge, lane == feature)
__global__ __launch_bounds__(256) void edge_scatter(const float* __restrict__ feat,
                                                    const int* __restrict__ src,
                                                    const int* __restrict__ dst,
                                                    const int* __restrict__ et,
                                                    float* __restrict__ S, int nEdges) {
    long long t = (long long)blockIdx.x * blockDim.x + threadIdx.x;
    int e = (int)(t >> 5);
    int f = (int)(t & 31);
    if (e >= nEdges) return;
    int s = src[e];
    int d = dst[e];
    int r = et[e];
    float v = feat[(size_t)s * FEAT + f];
    atomicAdd(S + ((size_t)r * N_NODES + d) * FEAT + f, v);
}

// ---------------- WMMA tile: acc(16x32) += S~(16x128) @ W(128x32) ----------------
// A 16x4 f32 layout: VGPR0 = K={k0 | k0+2}, VGPR1 = K={k0+1 | k0+3} (lane halves)
// B 4x16 f32 layout: VGPR0 = row {k0 | k0+2}, VGPR1 = row {k0+1 | k0+3}
__device__ __forceinline__ void wmma_tile_16x128(const float* __restrict__ S, int row0,
                                                 const float* __restrict__ W,
                                                 int half, int l16, v8f& acc0, v8f& acc1) {
#pragma unroll
    for (int k0 = 0; k0 < 128; k0 += 4) {
        const int r = k0 >> 5;               // relation block of K
        const int d = (k0 & 31) + 2 * half;  // feature index within relation
        const float* ap = S + ((size_t)r * N_NODES + (row0 + l16)) * FEAT + d;
        v2f a;
        a.x = ap[0];
        a.y = ap[1];
        const float* bp = W + (size_t)(k0 + 2 * half) * FEAT + l16;  // W viewed 128x32
        v2f b0, b1;
        b0.x = bp[0];
        b0.y = bp[FEAT];
        b1.x = bp[16];
        b1.y = bp[FEAT + 16];
        acc0 = __builtin_amdgcn_wmma_f32_16x16x4_f32(false, a, false, b0, (short)0, acc0,
                                                     false, false);
        acc1 = __builtin_amdgcn_wmma_f32_16x16x4_f32(false, a, false, b1, (short)0, acc1,
                                                     false, false);
    }
}

// ---------------- H = relu(S~ @ W + b), one wave per 16-node tile ----------------
__global__ __launch_bounds__(256) void transform_relu(const float* __restrict__ S,
                                                      const float* __restrict__ W,
                                                      const float* __restrict__ bias,
                                                      float* __restrict__ H) {
    const int wave = (blockIdx.x * blockDim.x + threadIdx.x) >> 5;
    const int lane = threadIdx.x & 31;
    const int half = lane >> 4;
    const int l16 = lane & 15;
    const int row0 = wave * 16;

    v8f acc0 = {};
    v8f acc1 = {};
    wmma_tile_16x128(S, row0, W, half, l16, acc0, acc1);

    const float bc0 = bias[l16];
    const float bc1 = bias[16 + l16];
    // C/D layout: VGPR j -> row j (lanes 0-15) / row j+8 (lanes 16-31), col = l16
#pragma unroll
    for (int j = 0; j < 8; ++j) {
        int row = row0 + j + half * 8;
        float v0 = acc0[j] + bc0;
        v0 = v0 > 0.f ? v0 : 0.f;
        float v1 = acc1[j] + bc1;
        v1 = v1 > 0.f ? v1 : 0.f;
        H[(size_t)row * FEAT + l16] = v0;
        H[(size_t)row * FEAT + 16 + l16] = v1;
    }
}

// ------- out[g] = mean_{32 nodes} relu(S~ @ W + b), one wave per group -------
__global__ __launch_bounds__(256) void transform_mean(const float* __restrict__ S,
                                                      const float* __restrict__ W,
                                                      const float* __restrict__ bias,
                                                      float* __restrict__ out) {
    const int g = (blockIdx.x * blockDim.x + threadIdx.x) >> 5;
    const int lane = threadIdx.x & 31;
    const int half = lane >> 4;
    const int l16 = lane & 15;

    const float bc0 = bias[l16];
    const float bc1 = bias[16 + l16];
    float cs0 = 0.f, cs1 = 0.f;

#pragma unroll
    for (int t = 0; t < 2; ++t) {
        const int row0 = g * 32 + t * 16;
        v8f acc0 = {};
        v8f acc1 = {};
        wmma_tile_16x128(S, row0, W, half, l16, acc0, acc1);
#pragma unroll
        for (int j = 0; j < 8; ++j) {
            float v0 = acc0[j] + bc0;
            cs0 += (v0 > 0.f ? v0 : 0.f);
            float v1 = acc1[j] + bc1;
            cs1 += (v1 > 0.f ? v1 : 0.f);
        }
    }
    // lanes L and L+16 hold the two row-halves of the same column: combine
    cs0 += __shfl_xor(cs0, 16);
    cs1 += __shfl_xor(cs1, 16);
    const float inv = 1.0f / 32.0f;
    if (half == 0)
        out[(size_t)g * FEAT + l16] = cs0 * inv;
    else
        out[(size_t)g * FEAT + 16 + l16] = cs1 * inv;
}

extern "C" void kernel_launch(void* const* d_in, const int* in_sizes, int n_in,
                              void* d_out, int out_size, void* d_ws, size_t ws_size,
                              hipStream_t stream) {
    const float* x = (const float*)d_in[0];
    const int* src = (const int*)d_in[1];
    const int* dst = (const int*)d_in[2];
    const int* et = (const int*)d_in[3];
    const float* W = (const float*)d_in[4];   // (4,32,32) == 128x32 row-major
    const float* b = (const float*)d_in[5];
    float* out = (float*)d_out;

    const int nEdges = in_sizes[1];

    // workspace layout: S = 4*N*32 f32 (256 MB), H = N*32 f32 (64 MB)
    float* S = (float*)d_ws;
    float* H = (float*)((char*)d_ws + (size_t)N_REL * N_NODES * FEAT * sizeof(float));

    const size_t n4 = (size_t)N_REL * N_NODES * FEAT / 4;
    const dim3 blk(256);
    const unsigned zeroGrid = (unsigned)((n4 + 255) / 256);
    const unsigned edgeGrid = (unsigned)(((long long)nEdges * 32 + 255) / 256);
    const unsigned xformGrid = N_NODES / 16 / 8;  // 8 waves per block, 16 rows per wave
    const unsigned meanGrid = N_NODES / 32 / 8;   // 8 waves per block, 1 group per wave

    // Layer 1: S = per-relation scatter of x; H = relu(S~ @ W + b)
    zero_f4<<<zeroGrid, blk, 0, stream>>>((float4*)S, n4);
    edge_scatter<<<edgeGrid, blk, 0, stream>>>(x, src, dst, et, S, nEdges);
    transform_relu<<<xformGrid, blk, 0, stream>>>(S, W, b, H);

    // Layer 2 fused with block-mean
    zero_f4<<<zeroGrid, blk, 0, stream>>>((float4*)S, n4);
    edge_scatter<<<edgeGrid, blk, 0, stream>>>(H, src, dst, et, S, nEdges);
    transform_mean<<<meanGrid, blk, 0, stream>>>(S, W, b, out);
}